// SelectionLayer_12008728559854
// MI455X (gfx1250) — compile-verified
//
#include <hip/hip_runtime.h>
#include <stdint.h>

// Problem constants (from reference): x is (64, 64, 96, 96) float32.
#define B_    64
#define C_    64
#define HW_   9216                 // 96*96
#define N_    (C_ * HW_)           // 589824 elements per batch
#define KSEL_ 294912u              // KEEP_PERCENT * N_  (top-k count per batch)
#define FIXL_ 1                    // FIX_LAYERS

#define TILE_FLOATS 4096           // floats staged to LDS per block in hist pass
#define BLOCKS_PER_BATCH (N_ / TILE_FLOATS)   // 144, exact

// Map float bits to a monotonically ordered unsigned key:
// larger float  <=>  larger key.
__device__ __forceinline__ uint32_t fkey(float f) {
    uint32_t u = __float_as_uint(f);
    uint32_t m = (u >> 31) ? 0xFFFFFFFFu : 0x80000000u;
    return u ^ m;
}

// ---------------------------------------------------------------------------
// Init: zero per-batch 256-bin histograms, reset prefix / remaining-k.
// ---------------------------------------------------------------------------
__global__ void sel_init(uint32_t* __restrict__ hist,
                         uint32_t* __restrict__ prefix,
                         uint32_t* __restrict__ kk) {
    const int b = blockIdx.x, t = threadIdx.x;
    hist[b * 256 + t] = 0u;
    if (t == 0) { prefix[b] = 0u; kk[b] = KSEL_; }
}

// ---------------------------------------------------------------------------
// Radix-select histogram pass. Each block stages a 16KB tile of x into LDS
// via the CDNA5 async data mover (global_load_async_to_lds_b128, ASYNCcnt),
// then histograms the 8-bit digit at `shift` for keys matching the current
// per-batch prefix.
// ---------------------------------------------------------------------------
__global__ void sel_hist(const float* __restrict__ x,
                         uint32_t* __restrict__ hist,
                         const uint32_t* __restrict__ prefix,
                         int shift) {
    __shared__ uint32_t lhist[256];
    __shared__ float    ldata[TILE_FLOATS];

    const int t = threadIdx.x;          // 0..255
    const int b = blockIdx.y;           // batch
    lhist[t] = 0u;

    const float* base = x + (size_t)b * N_ + (size_t)blockIdx.x * TILE_FLOATS;
    // Low 32 bits of the generic address of a __shared__ object are the LDS
    // byte offset (LDS aperture truncates to addr[31:0]).
    const uint32_t lbase = (uint32_t)(uintptr_t)(&ldata[0]);

    // 4 x 16B async copies per thread: 256 threads * 64B = 16KB tile.
    #pragma unroll
    for (int i = 0; i < 4; ++i) {
        const float* g    = base + (size_t)((i * 256 + t) * 4);
        const uint32_t lo = lbase + (uint32_t)((i * 256 + t) * 16);
        asm volatile("global_load_async_to_lds_b128 %0, %1, off"
                     :: "v"(lo), "v"(g) : "memory");
    }
    asm volatile("s_wait_asynccnt 0" ::: "memory");
    __syncthreads();   // lhist zeroing visible; async data landed

    const uint32_t pfx   = prefix[b];
    const uint32_t pmask = (uint32_t)(0xFFFFFFFFull << (shift + 8)); // 0 on pass 0

    #pragma unroll
    for (int i = 0; i < 4; ++i) {
        float4 v = ((const float4*)ldata)[i * 256 + t];
        uint32_t k0 = fkey(v.x), k1 = fkey(v.y), k2 = fkey(v.z), k3 = fkey(v.w);
        if ((k0 & pmask) == (pfx & pmask)) atomicAdd(&lhist[(k0 >> shift) & 255u], 1u);
        if ((k1 & pmask) == (pfx & pmask)) atomicAdd(&lhist[(k1 >> shift) & 255u], 1u);
        if ((k2 & pmask) == (pfx & pmask)) atomicAdd(&lhist[(k2 >> shift) & 255u], 1u);
        if ((k3 & pmask) == (pfx & pmask)) atomicAdd(&lhist[(k3 >> shift) & 255u], 1u);
    }
    __syncthreads();

    const uint32_t c = lhist[t];
    if (c) atomicAdd(&hist[b * 256 + t], c);
}

// ---------------------------------------------------------------------------
// Pick the digit bucket containing the K-th largest key; update prefix and
// remaining-k; re-zero this batch's histogram for the next pass.
// ---------------------------------------------------------------------------
__global__ void sel_pick(uint32_t* __restrict__ hist,
                         uint32_t* __restrict__ prefix,
                         uint32_t* __restrict__ kk,
                         int shift) {
    const int b = blockIdx.x, t = threadIdx.x;
    if (t == 0) {
        const uint32_t k = kk[b];
        uint32_t acc = 0u, dsel = 0u, krem = k;
        for (int d = 255; d >= 0; --d) {
            const uint32_t c = hist[b * 256 + d];
            if (acc + c >= k) { dsel = (uint32_t)d; krem = k - acc; break; }
            acc += c;
        }
        prefix[b] |= dsel << shift;
        kk[b] = krem;
    }
    __syncthreads();
    hist[b * 256 + t] = 0u;   // ready for next pass
}

// ---------------------------------------------------------------------------
// Final pass: per (b,h,w) position, 64-channel arg-max (first index wins on
// ties, matching lax.top_k), then masked write. Non-temporal stores keep x
// resident in the 192MB L2 across passes / graph replays.
// ---------------------------------------------------------------------------
__global__ void sel_out(const float* __restrict__ x,
                        float* __restrict__ out,
                        const uint32_t* __restrict__ prefix) {
    const int idx = blockIdx.x * 256 + threadIdx.x;   // 0 .. B*HW-1
    const int b   = idx / HW_;
    const int pos = idx - b * HW_;
    const uint32_t T = prefix[b];                     // key of K-th largest

    const float* xb = x   + (size_t)b * N_ + (size_t)pos;
    float*       ob = out + (size_t)b * N_ + (size_t)pos;

    float vals[C_];
    uint32_t bestKey = 0u;
    int bestC = 0;
    #pragma unroll
    for (int c = 0; c < C_; ++c) {
        const float v = xb[(size_t)c * HW_];
        vals[c] = v;
        const uint32_t k = fkey(v);
        if (k > bestKey) { bestKey = k; bestC = c; }  // strict > => lowest index wins
    }

    #pragma unroll
    for (int c = 0; c < C_; ++c) {
        const uint32_t k = fkey(vals[c]);
        const bool keep = (c < FIXL_) || (c == bestC) || (k >= T);
        __builtin_nontemporal_store(keep ? vals[c] : 0.0f, &ob[(size_t)c * HW_]);
    }
}

// ---------------------------------------------------------------------------
// Launch: init -> 4x(hist, pick) radix-select -> masked output.
// Workspace: 64*256 hist + 64 prefix + 64 kk u32 = 66,048 bytes.
// ---------------------------------------------------------------------------
extern "C" void kernel_launch(void* const* d_in, const int* in_sizes, int n_in,
                              void* d_out, int out_size, void* d_ws, size_t ws_size,
                              hipStream_t stream) {
    (void)in_sizes; (void)n_in; (void)out_size; (void)ws_size;
    const float* x = (const float*)d_in[0];
    float* out     = (float*)d_out;

    uint32_t* hist   = (uint32_t*)d_ws;       // 64 * 256
    uint32_t* prefix = hist + 64 * 256;       // 64
    uint32_t* kk     = prefix + 64;           // 64

    sel_init<<<64, 256, 0, stream>>>(hist, prefix, kk);

    for (int p = 0; p < 4; ++p) {
        const int shift = 24 - 8 * p;
        sel_hist<<<dim3(BLOCKS_PER_BATCH, B_), 256, 0, stream>>>(x, hist, prefix, shift);
        sel_pick<<<64, 256, 0, stream>>>(hist, prefix, kk, shift);
    }

    sel_out<<<(B_ * HW_) / 256, 256, 0, stream>>>(x, out, prefix);
}